// GCNModel_3332894622180
// MI455X (gfx1250) — compile-verified
//
#include <hip/hip_runtime.h>
#include <hip/hip_bf16.h>

#define F_IN 1433
#define HID  16
#define NCLS 7
#define KPAD 1436          // F_IN rounded up to a multiple of 4 (zero-padded in LDS)

typedef __attribute__((ext_vector_type(2))) float v2f;
typedef __attribute__((ext_vector_type(8))) float v8f;
typedef __attribute__((ext_vector_type(4))) int   v4i;

#if defined(__gfx1250__) && __has_builtin(__builtin_amdgcn_global_load_async_to_lds_b128)
#define HAVE_ASYNC_LDS 1
#endif

// ---------------------------------------------------------------- utilities
__global__ void zero_f32(float* __restrict__ p, long long n) {
  long long i = (long long)blockIdx.x * blockDim.x + threadIdx.x;
  if (i < n) p[i] = 0.0f;
}

__global__ void deg_count(const int* __restrict__ dst, float* __restrict__ deg, int E) {
  int e = blockIdx.x * blockDim.x + threadIdx.x;
  if (e < E) atomicAdd(&deg[dst[e]], 1.0f);
}

// deg -> 1/sqrt(deg + 1)   (+1 accounts for the self loop)
__global__ void deg_to_dis(float* __restrict__ deg, int n) {
  int i = blockIdx.x * blockDim.x + threadIdx.x;
  if (i < n) deg[i] = rsqrtf(deg[i] + 1.0f);
}

// ----------------------------------------------------- GEMM1: h1 = x @ W1
// One wave per 16x16 output tile, V_WMMA_F32_16X16X4_F32 over K.
// W1 (1433x16 f32 = 92KB) is staged into LDS once per block via the CDNA5
// async global->LDS path (ASYNCcnt), zero-padded to KPAD rows so the K loop
// is uniform and the final WMMA step needs no guarded B loads.
__global__ void __launch_bounds__(256)
gemm1_wmma(const float* __restrict__ x, const float* __restrict__ W,
           float* __restrict__ h, int n) {
  __shared__ float Ws[KPAD * HID];          // 91,904 B of the 320KB/WGP LDS

  const int tid = threadIdx.x;

  // zero the pad rows (k = F_IN .. KPAD-1): (KPAD-F_IN)*HID = 48 slots
  if (tid < (KPAD - F_IN) * HID) Ws[F_IN * HID + tid] = 0.0f;

#if defined(HAVE_ASYNC_LDS)
  // async-copy the valid 1433*16 floats as b128 chunks (5732 vec4 copies)
  for (int i = tid; i < (F_IN * HID) / 4; i += 256) {
    __builtin_amdgcn_global_load_async_to_lds_b128(
        (__attribute__((address_space(1))) v4i*)(const_cast<float*>(W) + (size_t)i * 4),
        (__attribute__((address_space(3))) v4i*)(Ws + (size_t)i * 4),
        0, 0);
  }
#if __has_builtin(__builtin_amdgcn_s_wait_asynccnt)
  __builtin_amdgcn_s_wait_asynccnt(0);
#else
  asm volatile("s_wait_asynccnt 0x0" ::: "memory");
#endif
#else
  for (int i = tid; i < (F_IN * HID) / 4; i += 256) {
    const float4 v = ((const float4*)W)[i];
    ((float4*)Ws)[i] = v;
  }
#endif
  __syncthreads();                          // all waves reach this before any exit

  const int lane    = tid & 31;
  const int tile    = blockIdx.x * 8 + (tid >> 5);
  const int rowBase = tile << 4;
  if (rowBase >= n) return;                 // uniform per wave -> EXEC all-1s at WMMA

  const int m  = lane & 15;                 // output column index
  const int kh = (lane >> 4) << 1;          // 0 (lanes 0-15) or 2 (lanes 16-31)
  int row = rowBase + m;
  if (row >= n) row = n - 1;                // clamp loads for a (non-existent) tail tile
  const float* xrow = x + (size_t)row * F_IN;

  v8f acc = {};
  int k = 0;
  for (; k < F_IN - 1; k += 4) {            // 358 full steps: k+kh+1 <= 1432 always
    v2f a, b;
    a.x = xrow[k + kh];
    a.y = xrow[k + kh + 1];
    b.x = Ws[(k + kh)     * HID + m];
    b.y = Ws[(k + kh + 1) * HID + m];
    acc = __builtin_amdgcn_wmma_f32_16x16x4_f32(false, a, false, b,
                                                (short)0, acc, false, false);
  }
  {                                         // tail step k = 1432..1435 (B pad is 0)
    v2f a = {0.0f, 0.0f}, b;
    const int k0 = k + kh, k1 = k + kh + 1;
    if (k0 < F_IN) a.x = xrow[k0];
    if (k1 < F_IN) a.y = xrow[k1];
    b.x = Ws[k0 * HID + m];
    b.y = Ws[k1 * HID + m];
    acc = __builtin_amdgcn_wmma_f32_16x16x4_f32(false, a, false, b,
                                                (short)0, acc, false, false);
  }

  // C/D layout: vgpr i -> row M = i + 8*(lane>=16), col N = lane&15
  const int rHalf = (lane >> 4) << 3;
  if (rowBase + 15 < n) {                   // common case: whole tile in range
    #pragma unroll
    for (int i = 0; i < 8; ++i)
      h[(size_t)(rowBase + rHalf + i) * HID + m] = acc[i];
  } else {
    #pragma unroll
    for (int i = 0; i < 8; ++i) {
      int r = rowBase + rHalf + i;
      if (r < n) h[(size_t)r * HID + m] = acc[i];
    }
  }
}

// ------------------------------------------- GEMM2: h2 = relu1 @ W2 (7 cols)
// K = 16 -> exactly 4 WMMA steps; W2 zero-padded to 16 columns, h2 stride 8.
__global__ void __launch_bounds__(256)
gemm2_wmma(const float* __restrict__ h, const float* __restrict__ W2,
           float* __restrict__ h2, int n) {
  const int lane    = threadIdx.x & 31;
  const int tile    = blockIdx.x * 8 + (threadIdx.x >> 5);
  const int rowBase = tile << 4;
  if (rowBase >= n) return;

  const int m  = lane & 15;
  const int kh = (lane >> 4) << 1;
  int row = rowBase + m;
  if (row >= n) row = n - 1;
  const float* hrow = h + (size_t)row * HID;

  v8f acc = {};
  #pragma unroll
  for (int k = 0; k < HID; k += 4) {
    v2f a, b;
    a.x = hrow[k + kh];
    a.y = hrow[k + kh + 1];
    b.x = (m < NCLS) ? W2[(size_t)(k + kh)     * NCLS + m] : 0.0f;
    b.y = (m < NCLS) ? W2[(size_t)(k + kh + 1) * NCLS + m] : 0.0f;
    acc = __builtin_amdgcn_wmma_f32_16x16x4_f32(false, a, false, b,
                                                (short)0, acc, false, false);
  }

  const int rHalf = (lane >> 4) << 3;
  if (m < 8) {                              // store cols 0..7 (col 7 is exact 0)
    if (rowBase + 15 < n) {
      #pragma unroll
      for (int i = 0; i < 8; ++i)
        h2[(size_t)(rowBase + rHalf + i) * 8 + m] = acc[i];
    } else {
      #pragma unroll
      for (int i = 0; i < 8; ++i) {
        int r = rowBase + rHalf + i;
        if (r < n) h2[(size_t)r * 8 + m] = acc[i];
      }
    }
  }
}

// --------------------------------------------- edge scatter (16 lanes/edge)
__global__ void agg_edges_16(const int* __restrict__ src, const int* __restrict__ dst,
                             const float* __restrict__ dis, const float* __restrict__ h,
                             float* __restrict__ agg, int E) {
  long long t = (long long)blockIdx.x * blockDim.x + threadIdx.x;
  int e = (int)(t >> 4);
  if (e >= E) return;
  int j = (int)(t & 15);
  int s = src[e], d = dst[e];
  float coef = dis[s] * dis[d];
  atomicAdd(agg + (size_t)d * HID + j, h[(size_t)s * HID + j] * coef);
}

// ---------------------------------------------- edge scatter (8 lanes/edge)
__global__ void agg_edges_8(const int* __restrict__ src, const int* __restrict__ dst,
                            const float* __restrict__ dis, const float* __restrict__ h,
                            float* __restrict__ agg, int E) {
  long long t = (long long)blockIdx.x * blockDim.x + threadIdx.x;
  int e = (int)(t >> 3);
  if (e >= E) return;
  int j = (int)(t & 7);                     // col 7 of h is 0 -> harmless
  int s = src[e], d = dst[e];
  float coef = dis[s] * dis[d];
  atomicAdd(agg + (size_t)d * 8 + j, h[(size_t)s * 8 + j] * coef);
}

// ---------------- layer-1 finalize: + self-loop + bias, ReLU (in-place in h1)
__global__ void finalize1(const float* __restrict__ agg, const float* __restrict__ dis,
                          const float* __restrict__ b1, float* __restrict__ h, int n) {
  long long t = (long long)blockIdx.x * blockDim.x + threadIdx.x;
  int i = (int)(t >> 4);
  if (i >= n) return;
  int j = (int)(t & 15);
  float self = dis[i] * dis[i];
  float v = agg[(size_t)i * HID + j] + h[(size_t)i * HID + j] * self + b1[j];
  h[(size_t)i * HID + j] = fmaxf(v, 0.0f);
}

// -------- layer-2 finalize: + self-loop + bias, ReLU, log_softmax over 7 cls
__global__ void finalize2(const float* __restrict__ agg, const float* __restrict__ h2,
                          const float* __restrict__ dis, const float* __restrict__ b2,
                          float* __restrict__ out, int n) {
  int i = blockIdx.x * blockDim.x + threadIdx.x;
  if (i >= n) return;
  float self = dis[i] * dis[i];
  float v[NCLS];
  float mx = -3.402823466e38f;
  #pragma unroll
  for (int j = 0; j < NCLS; ++j) {
    float t = agg[(size_t)i * 8 + j] + h2[(size_t)i * 8 + j] * self + b2[j];
    t = fmaxf(t, 0.0f);
    v[j] = t;
    mx = fmaxf(mx, t);
  }
  float s = 0.0f;
  #pragma unroll
  for (int j = 0; j < NCLS; ++j) s += __expf(v[j] - mx);
  float lse = mx + __logf(s);
  #pragma unroll
  for (int j = 0; j < NCLS; ++j) out[(size_t)i * NCLS + j] = v[j] - lse;
}

// --------------------------------------------------------------------------
extern "C" void kernel_launch(void* const* d_in, const int* in_sizes, int n_in,
                              void* d_out, int out_size, void* d_ws, size_t ws_size,
                              hipStream_t stream) {
  const float* x  = (const float*)d_in[0];
  const int*   ei = (const int*)d_in[1];
  const float* W1 = (const float*)d_in[2];
  const float* b1 = (const float*)d_in[3];
  const float* W2 = (const float*)d_in[4];
  const float* b2 = (const float*)d_in[5];
  float* out = (float*)d_out;

  const int n = in_sizes[0] / F_IN;   // 100000
  const int E = in_sizes[1] / 2;      // 3200000
  const int* src = ei;
  const int* dst = ei + E;

  // workspace layout (floats): [deg|agg1|agg2 | h1|h2]  (first 3 zero-initialized)
  float* ws   = (float*)d_ws;
  float* deg  = ws;                          // n        (becomes deg^{-1/2})
  float* agg1 = deg  + n;                    // n*16
  float* agg2 = agg1 + (size_t)n * HID;      // n*8
  float* h1   = agg2 + (size_t)n * 8;        // n*16
  float* h2   = h1   + (size_t)n * HID;      // n*8

  const long long zeroN = (long long)n * (1 + HID + 8);
  zero_f32<<<(int)((zeroN + 255) / 256), 256, 0, stream>>>(ws, zeroN);

  deg_count <<<(E + 255) / 256, 256, 0, stream>>>(dst, deg, E);
  deg_to_dis<<<(n + 255) / 256, 256, 0, stream>>>(deg, n);

  const int tiles   = (n + 15) / 16;
  const int gBlocks = (tiles + 7) / 8;       // 8 waves per 256-thread block

  gemm1_wmma<<<gBlocks, 256, 0, stream>>>(x, W1, h1, n);

  const long long t16 = (long long)E * 16;
  agg_edges_16<<<(int)((t16 + 255) / 256), 256, 0, stream>>>(src, dst, deg, h1, agg1, E);

  const long long n16 = (long long)n * 16;
  finalize1<<<(int)((n16 + 255) / 256), 256, 0, stream>>>(agg1, deg, b1, h1, n);

  gemm2_wmma<<<gBlocks, 256, 0, stream>>>(h1, W2, h2, n);

  const long long t8 = (long long)E * 8;
  agg_edges_8<<<(int)((t8 + 255) / 256), 256, 0, stream>>>(src, dst, deg, h2, agg2, E);

  finalize2<<<(n + 255) / 256, 256, 0, stream>>>(agg2, h2, deg, b2, out, n);
}